// QuantumFeedForward_91199335563771
// MI455X (gfx1250) — compile-verified
//
#include <hip/hip_runtime.h>

// ---------------------------------------------------------------------------
// QuantumFeedForward for MI455X (gfx1250), reformulated as 3 WMMA GEMMs.
//
//  out = ( |M @ psi0(tanh(x@W_in^T+b_in)*pi)|^2 ) @ (W_out@Z)^T + b_out
//
//  M   : fixed 16x16 complex unitary (Rot layers + CNOT rings), from weights
//  psi0: real product state from per-sample RY angles
//  G   : (64x16) = W_out @ Z   (Z_w expectation folded into output proj)
//
//  Roofline: 64 MB traffic / 23.3 TB/s ~= 2.8 us; ~470 MFLOP of f32 WMMA
//  (40 x V_WMMA_F32_16X16X4_F32 per 16-sample wave tile) -> memory bound.
//
//  All B-operand fragments come from a precomputed workspace (zero-padded
//  where needed) so the hot loops contain only unconditional b64 loads and
//  WMMA. tanh is computed branchlessly via v_exp_f32 to avoid OCML's
//  EXEC-predicated range reduction.
// ---------------------------------------------------------------------------

typedef __attribute__((ext_vector_type(2))) float v2f;
typedef __attribute__((ext_vector_type(8))) float v8f;

#define N_SAMPLES_PER_WAVE 16
#define WAVES_PER_BLOCK 8

// ws layout (floats):
//   [0   ..255 ]  Mre[j][i]   (row j = output amp index, col i = input index)
//   [256 ..511 ]  Mim[j][i]
//   [512 ..1535]  G[d][i]     (64 x 16)
//   [1536..2559]  B1t[n][k]   (16 x 64) = W_in zero-padded to 16 rows
__global__ void qffn_setup(const float* __restrict__ weights,
                           const float* __restrict__ W_in,
                           const float* __restrict__ W_out,
                           float* __restrict__ ws) {
  __shared__ float Mr[16][16];  // [col][row]
  __shared__ float Mi[16][16];
  const int t = threadIdx.x;

  if (t < 16) {
    // Thread t owns column t of M. Build M = prod(CNOTring * RotLayer).
    for (int r = 0; r < 16; ++r) {
      Mr[t][r] = (r == t) ? 1.0f : 0.0f;
      Mi[t][r] = 0.0f;
    }
    for (int layer = 0; layer < 3; ++layer) {
      for (int w = 0; w < 4; ++w) {
        const float phi   = weights[(layer * 4 + w) * 3 + 0];
        const float theta = weights[(layer * 4 + w) * 3 + 1];
        const float omega = weights[(layer * 4 + w) * 3 + 2];
        const float ct = cosf(0.5f * theta), st = sinf(0.5f * theta);
        const float ap = -0.5f * (phi + omega);   // U00 = ct*e^{i*ap}
        const float am =  0.5f * (phi - omega);   // U01 = -st*e^{i*am}
        const float c_ap = cosf(ap), s_ap = sinf(ap);
        const float c_am = cosf(am), s_am = sinf(am);
        const float U00r =  ct * c_ap, U00i =  ct * s_ap;
        const float U01r = -st * c_am, U01i = -st * s_am;
        const float U10r =  st * c_am, U10i = -st * s_am;
        const float U11r =  ct * c_ap, U11i = -ct * s_ap;
        const int mask = 8 >> w;  // wire 0 = MSB of flattened (2,2,2,2) index
        for (int r = 0; r < 16; ++r) {
          if (r & mask) continue;
          const int r1 = r | mask;
          const float x0r = Mr[t][r],  x0i = Mi[t][r];
          const float x1r = Mr[t][r1], x1i = Mi[t][r1];
          Mr[t][r]  = U00r * x0r - U00i * x0i + U01r * x1r - U01i * x1i;
          Mi[t][r]  = U00r * x0i + U00i * x0r + U01r * x1i + U01i * x1r;
          Mr[t][r1] = U10r * x0r - U10i * x0i + U11r * x1r - U11i * x1i;
          Mi[t][r1] = U10r * x0i + U10i * x0r + U11r * x1i + U11i * x1r;
        }
      }
      // CNOT ring: state'[r] = state[sigma(r)] -> permute rows.
      for (int w = 0; w < 4; ++w) {
        const int cm = 8 >> w, tm = 8 >> ((w + 1) & 3);
        for (int r = 0; r < 16; ++r) {
          if ((r & cm) && !(r & tm)) {
            const int r1 = r | tm;
            float tmp;
            tmp = Mr[t][r]; Mr[t][r] = Mr[t][r1]; Mr[t][r1] = tmp;
            tmp = Mi[t][r]; Mi[t][r] = Mi[t][r1]; Mi[t][r1] = tmp;
          }
        }
      }
    }
  }
  __syncthreads();

  if (t < 16) {
    // Mre[j][i] = Mr[i][j] (we stored columns); here i == t.
    for (int j = 0; j < 16; ++j) {
      ws[j * 16 + t]       = Mr[t][j];
      ws[256 + j * 16 + t] = Mi[t][j];
    }
  }
  // G[d][i] = sum_w W_out[d][w] * (1 - 2*bit_w(i)),  bit_w(i) = (i >> (3-w)) & 1
  if (t < 64) {
    const float w0 = W_out[t * 4 + 0], w1 = W_out[t * 4 + 1];
    const float w2 = W_out[t * 4 + 2], w3 = W_out[t * 4 + 3];
    for (int i = 0; i < 16; ++i) {
      float g = ((i & 8) ? -w0 : w0) + ((i & 4) ? -w1 : w1) +
                ((i & 2) ? -w2 : w2) + ((i & 1) ? -w3 : w3);
      ws[512 + t * 16 + i] = g;
    }
    // Zero-padded B1^T: rows 0..3 = W_in, rows 4..15 = 0.  (t indexes k.)
    for (int n = 0; n < 16; ++n) {
      ws[1536 + n * 64 + t] = (n < 4) ? W_in[n * 64 + t] : 0.0f;
    }
  }
}

__global__ __launch_bounds__(256) void qffn_main(
    const float* __restrict__ x, const float* __restrict__ b_in,
    const float* __restrict__ ws, const float* __restrict__ b_out,
    float* __restrict__ out) {
  // Per-wave LDS scratch (waves own disjoint slices).
  __shared__ float cs_lds[WAVES_PER_BLOCK][16][4][2];  // (c,s) per sample/wire
  __shared__ float p_lds[WAVES_PER_BLOCK][16][17];     // C->A transpose of p

  const int lane = threadIdx.x & 31;
  const int wv   = threadIdx.x >> 5;
  const int n    = lane & 15;   // WMMA N / A-layout M index
  const int hi   = lane >> 4;   // lane half selector
  const int tile = blockIdx.x * WAVES_PER_BLOCK + wv;
  const int s0   = tile * N_SAMPLES_PER_WAVE;

  // --------------------------------------------------------------------
  // GEMM1: C1 = X(16x64) @ B1(64x16), B1 = zero-padded W_in^T (from ws).
  // A f32 16x4 layout: lane -> M = lane%16, K = 4*kk + 2*hi + vgpr.
  // --------------------------------------------------------------------
  v8f c1 = {};
  const float* xrow  = x + (size_t)(s0 + n) * 64;
  const float* b1row = ws + 1536 + n * 64;
#pragma unroll
  for (int kk = 0; kk < 16; ++kk) {
    const int k = 4 * kk + 2 * hi;
    v2f a, b;
    a.x = xrow[k];
    a.y = xrow[k + 1];
    b.x = b1row[k];
    b.y = b1row[k + 1];
    c1 = __builtin_amdgcn_wmma_f32_16x16x4_f32(false, a, false, b,
                                               (short)0, c1, false, false);
  }

  // --------------------------------------------------------------------
  // Elementwise: angle = tanh(.+b_in)*pi/2, half-angle trig -> LDS.
  // Branchless tanh: 1 - 2/(e^{2x}+1)  (overflow-safe: e->inf => 1).
  // C layout: element (m = r + 8*hi, col = n).
  // --------------------------------------------------------------------
  const float bi = b_in[n & 3];  // branchless; only n<4 lanes matter
#pragma unroll
  for (int r = 0; r < 8; ++r) {
    if (n < 4) {
      const int m = r + 8 * hi;
      const float e  = __expf(2.0f * (c1[r] + bi));
      const float th = 1.0f - __fdividef(2.0f, e + 1.0f);
      const float ang = th * 1.57079632679489662f;  // * pi/2
      cs_lds[wv][m][n][0] = __cosf(ang);
      cs_lds[wv][m][n][1] = __sinf(ang);
    }
  }
  __syncthreads();

  // --------------------------------------------------------------------
  // psi0 (real product state) directly in A-fragment layout, K = 16.
  // --------------------------------------------------------------------
  float cw[4], sw[4];
#pragma unroll
  for (int w = 0; w < 4; ++w) {
    cw[w] = cs_lds[wv][n][w][0];
    sw[w] = cs_lds[wv][n][w][1];
  }
  auto psi0 = [&](int i) {
    const float p01 = ((i & 8) ? sw[0] : cw[0]) * ((i & 4) ? sw[1] : cw[1]);
    const float p23 = ((i & 2) ? sw[2] : cw[2]) * ((i & 1) ? sw[3] : cw[3]);
    return p01 * p23;
  };
  v2f a2[4];
#pragma unroll
  for (int kk = 0; kk < 4; ++kk) {
    const int i0 = 4 * kk + 2 * hi;
    a2[kk].x = psi0(i0);
    a2[kk].y = psi0(i0 + 1);
  }

  // --------------------------------------------------------------------
  // GEMM2: Psi_{re,im} = psi0 @ M^T.  B[k=i][n=j] = M{re,im}[j][i].
  // --------------------------------------------------------------------
  v8f cre = {}, cim = {};
#pragma unroll
  for (int kk = 0; kk < 4; ++kk) {
    const int k = 4 * kk + 2 * hi;
    v2f br, bim;
    br.x  = ws[n * 16 + k];
    br.y  = ws[n * 16 + k + 1];
    bim.x = ws[256 + n * 16 + k];
    bim.y = ws[256 + n * 16 + k + 1];
    cre = __builtin_amdgcn_wmma_f32_16x16x4_f32(false, a2[kk], false, br,
                                                (short)0, cre, false, false);
    cim = __builtin_amdgcn_wmma_f32_16x16x4_f32(false, a2[kk], false, bim,
                                                (short)0, cim, false, false);
  }

  // p = |psi|^2, transpose C-layout -> A-layout through LDS.
#pragma unroll
  for (int r = 0; r < 8; ++r) {
    const int m = r + 8 * hi;
    p_lds[wv][m][n] = cre[r] * cre[r] + cim[r] * cim[r];
  }
  __syncthreads();

  v2f a3[4];
#pragma unroll
  for (int kk = 0; kk < 4; ++kk) {
    const int k = 4 * kk + 2 * hi;
    a3[kk].x = p_lds[wv][n][k];
    a3[kk].y = p_lds[wv][n][k + 1];
  }

  // --------------------------------------------------------------------
  // GEMM3: out = P @ G^T + b_out.  B[k=i][n] = G[nt*16+n][i].
  // --------------------------------------------------------------------
  float* orow = out + (size_t)s0 * 64;
#pragma unroll
  for (int nt = 0; nt < 4; ++nt) {
    const int d = nt * 16 + n;
    v8f c3 = {};
#pragma unroll
    for (int kk = 0; kk < 4; ++kk) {
      const int k = 4 * kk + 2 * hi;
      v2f b;
      b.x = ws[512 + d * 16 + k];
      b.y = ws[512 + d * 16 + k + 1];
      c3 = __builtin_amdgcn_wmma_f32_16x16x4_f32(false, a3[kk], false, b,
                                                 (short)0, c3, false, false);
    }
    const float bias = b_out[d];
#pragma unroll
    for (int r = 0; r < 8; ++r) {
      const int m = r + 8 * hi;
      orow[(size_t)m * 64 + d] = c3[r] + bias;  // 64B-contiguous per half-wave
    }
  }
}

extern "C" void kernel_launch(void* const* d_in, const int* in_sizes, int n_in,
                              void* d_out, int out_size, void* d_ws, size_t ws_size,
                              hipStream_t stream) {
  const float* x       = (const float*)d_in[0];
  const float* W_in    = (const float*)d_in[1];
  const float* b_in    = (const float*)d_in[2];
  const float* weights = (const float*)d_in[3];
  const float* W_out   = (const float*)d_in[4];
  const float* b_out   = (const float*)d_in[5];
  float* out = (float*)d_out;
  float* ws  = (float*)d_ws;

  // Materialize M (16x16 complex), G = W_out @ Z, and zero-padded W_in^T
  // into ws (fully rewritten each call -> deterministic).
  hipLaunchKernelGGL(qffn_setup, dim3(1), dim3(64), 0, stream,
                     weights, W_in, W_out, ws);

  const int nsamples = in_sizes[0] / 64;                       // B*S
  const int tiles    = nsamples / N_SAMPLES_PER_WAVE;          // 8192
  const int blocks   = tiles / WAVES_PER_BLOCK;                // 1024
  hipLaunchKernelGGL(qffn_main, dim3(blocks), dim3(256), 0, stream,
                     x, b_in, ws, b_out, out);
}